// ResGatedGraphConv_936302871049
// MI455X (gfx1250) — compile-verified
//
#include <hip/hip_runtime.h>
#include <hip/hip_bf16.h>

typedef __attribute__((ext_vector_type(2))) float v2f;
typedef __attribute__((ext_vector_type(8))) float v8f;

#define CIN 64
#define COUT 64

// ---------------------------------------------------------------------------
// Kernel 1: per-node linears via WMMA f32 16x16x4.
//   sg = x@W_sg + b_sg ; dg = x@W_dg + b_dg ; m = x@W_m + b_m   -> workspace
//   out = x@W_r + b_r                                            -> d_out
// Block = 128 threads = 4 waves; block handles 16 rows; wave w handles output
// columns [16w, 16w+16). K=64 -> 16 WMMA(16x16x4) per output tile.
// ---------------------------------------------------------------------------
__global__ __launch_bounds__(128) void node_gemm_wmma(
    const float* __restrict__ x,
    const float* __restrict__ W_sg, const float* __restrict__ b_sg,
    const float* __restrict__ W_dg, const float* __restrict__ b_dg,
    const float* __restrict__ W_m,  const float* __restrict__ b_m,
    const float* __restrict__ W_r,  const float* __restrict__ b_r,
    float* __restrict__ sg, float* __restrict__ dg, float* __restrict__ mout,
    float* __restrict__ out, int N)
{
    const int lane = threadIdx.x & 31;
    const int wave = threadIdx.x >> 5;
    const int row0 = blockIdx.x * 16;
    const int mrow = lane & 15;
    const int hi   = lane >> 4;              // 0: K pair {0,1}; 1: K pair {2,3}
    const bool full = (row0 + 16 <= N);

    // A fragments: lanes 0-15 hold K pairs {4k,4k+1}; lanes 16-31 {4k+2,4k+3}.
    int arow = row0 + mrow; if (arow >= N) arow = N - 1;
    const float2* xr = (const float2*)(x + (size_t)arow * CIN);
    v2f a[16];
#pragma unroll
    for (int kk = 0; kk < 16; ++kk) {
        float2 t = xr[2 * kk + hi];
        a[kk].x = t.x; a[kk].y = t.y;
    }

    const int col = wave * 16 + (lane & 15);
    const float* Ws[4]   = {W_sg, W_dg, W_m, W_r};
    const float* bias[4] = {b_sg, b_dg, b_m, b_r};
    float*       Os[4]   = {sg, dg, mout, out};

#pragma unroll
    for (int mtx = 0; mtx < 4; ++mtx) {
        const float* W = Ws[mtx];
        v8f c = {};
#pragma unroll
        for (int kk = 0; kk < 16; ++kk) {
            const int k0 = kk * 4 + hi * 2;
            v2f b;
            b.x = W[(size_t)k0 * COUT + col];
            b.y = W[(size_t)(k0 + 1) * COUT + col];
            c = __builtin_amdgcn_wmma_f32_16x16x4_f32(
                    false, a[kk], false, b, (short)0, c, false, false);
        }
        const float bv = bias[mtx][col];
        float* O = Os[mtx] + (size_t)(row0 + hi * 8) * COUT + col;
        if (full) {
            // straight-line stores: row M = r + hi*8, stride COUT floats
#pragma unroll
            for (int r = 0; r < 8; ++r)
                O[(size_t)r * COUT] = c[r] + bv;
        } else {
#pragma unroll
            for (int r = 0; r < 8; ++r) {
                if (row0 + hi * 8 + r < N) O[(size_t)r * COUT] = c[r] + bv;
            }
        }
    }
}

// ---------------------------------------------------------------------------
// Kernel 2: edge phase. One HALF-WAVE (16 lanes) per edge; each lane owns 4
// channels via float4 (3x global_load_b128 per lane). Scatter with hardware
// f32 atomics (inline asm guarantees global_atomic_add_f32, device scope —
// no CAS-loop fallback). sg/dg/m (25.6 MB each) stay L2-resident (192 MB L2).
// ---------------------------------------------------------------------------
__device__ __forceinline__ void atomic_add_f32_dev(float* p, float v) {
    asm volatile("global_atomic_add_f32 %0, %1, off scope:SCOPE_DEV"
                 :: "v"(p), "v"(v) : "memory");
}

__global__ __launch_bounds__(256) void edge_scatter(
    const float* __restrict__ sg, const float* __restrict__ dg,
    const float* __restrict__ m,  const int* __restrict__ ei,
    float* __restrict__ out, int E)
{
    const int l16  = threadIdx.x & 15;       // lane within half-wave
    const int half = threadIdx.x >> 4;       // half-wave id in block: 0..15
    const int e    = blockIdx.x * 16 + half;
    if (e >= E) return;

    const int row = ei[e];          // source node
    const int col = ei[E + e];      // destination node

    const float4 s  = ((const float4*)(sg + (size_t)row * COUT))[l16];
    const float4 d  = ((const float4*)(dg + (size_t)col * COUT))[l16];
    const float4 mv = ((const float4*)(m  + (size_t)row * COUT))[l16];

    const float g0 = 1.0f / (1.0f + __expf(-(s.x + d.x)));
    const float g1 = 1.0f / (1.0f + __expf(-(s.y + d.y)));
    const float g2 = 1.0f / (1.0f + __expf(-(s.z + d.z)));
    const float g3 = 1.0f / (1.0f + __expf(-(s.w + d.w)));

    float* o = out + (size_t)col * COUT + l16 * 4;
    atomic_add_f32_dev(o + 0, g0 * mv.x);
    atomic_add_f32_dev(o + 1, g1 * mv.y);
    atomic_add_f32_dev(o + 2, g2 * mv.z);
    atomic_add_f32_dev(o + 3, g3 * mv.w);
}

extern "C" void kernel_launch(void* const* d_in, const int* in_sizes, int n_in,
                              void* d_out, int out_size, void* d_ws, size_t ws_size,
                              hipStream_t stream)
{
    const float* x    = (const float*)d_in[0];
    const int*   ei   = (const int*)  d_in[1];
    const float* W_sg = (const float*)d_in[2];
    const float* b_sg = (const float*)d_in[3];
    const float* W_dg = (const float*)d_in[4];
    const float* b_dg = (const float*)d_in[5];
    const float* W_m  = (const float*)d_in[6];
    const float* b_m  = (const float*)d_in[7];
    const float* W_r  = (const float*)d_in[8];
    const float* b_r  = (const float*)d_in[9];
    float* out = (float*)d_out;

    const int N = in_sizes[0] / CIN;
    const int E = in_sizes[1] / 2;

    float* sg = (float*)d_ws;
    float* dg = sg + (size_t)N * COUT;
    float* mm = dg + (size_t)N * COUT;

    node_gemm_wmma<<<(N + 15) / 16, 128, 0, stream>>>(
        x, W_sg, b_sg, W_dg, b_dg, W_m, b_m, W_r, b_r, sg, dg, mm, out, N);

    edge_scatter<<<(E + 15) / 16, 256, 0, stream>>>(sg, dg, mm, ei, out, E);
}